// LowLevelRNN_6932077216194
// MI455X (gfx1250) — compile-verified
//
#include <hip/hip_runtime.h>
#include <math.h>

typedef float v2f __attribute__((ext_vector_type(2)));
typedef float v8f __attribute__((ext_vector_type(8)));

#define T_STEPS 8192
#define X_DIM   1024
#define H_DIM   2048
#define Y_DIM   1024

// Recurrence partitioning: 256 persistent blocks, 8 columns each, 64KB LDS weights
#define NWG 256
#define CPB 8      // columns per block (one wave per column)
#define RTH 256    // threads per recurrence block (8 waves)

// GEMM tiling
#define BM 128
#define BN 64
#define BK 32
#define GTH 256    // threads per GEMM block (8 waves)

// ---------------------------------------------------------------------------
// gfx1250 async global->LDS DMA (ASYNCcnt-tracked).
// ---------------------------------------------------------------------------
__device__ __forceinline__ void async_copy_b128(void* lds, const void* gptr)
{
    unsigned l32 = (unsigned)(size_t)lds;   // low 32 bits of flat = LDS offset
    asm volatile("global_load_async_to_lds_b128 %0, %1, off"
                 :: "v"(l32), "v"(gptr)
                 : "memory");
}

__device__ __forceinline__ void wait_async0()
{
    asm volatile("s_wait_asynccnt 0x0" ::: "memory");
}

// ---------------------------------------------------------------------------
// WMMA f32 GEMM: C[M,N] = A[M,K] * B[K,N] (+ bias[N] if bias != nullptr)
// Block computes a BMxBN tile; A/B K-chunks double-buffered in LDS via
// async DMA; 8 waves each own a 16-row strip and do 4 WMMA tiles along N.
// ---------------------------------------------------------------------------
__global__ __launch_bounds__(GTH)
void gemm_f32_wmma(const float* __restrict__ A, const float* __restrict__ B,
                   const float* __restrict__ bias, float* __restrict__ C,
                   int M, int N, int K)
{
    __shared__ __align__(16) float As[2][BM * BK];   // row-major [row][k]
    __shared__ __align__(16) float Bs[2][BK * BN];   // row-major [k][col]

    const int tid  = threadIdx.x;
    const int lane = tid & 31;
    const int wave = tid >> 5;                       // 0..7: 16-row strip

    const int nb  = N / BN;
    const int tm0 = (blockIdx.x / nb) * BM;
    const int tn0 = (blockIdx.x % nb) * BN;

    auto stage = [&](int buf, int kb) {
        #pragma unroll
        for (int i = 0; i < (BM * BK) / (4 * GTH); ++i) {
            const int cid = tid + i * GTH;
            const int row = cid >> 3;
            const int c4  = cid & 7;
            async_copy_b128(&As[buf][row * BK + c4 * 4],
                            A + (size_t)(tm0 + row) * K + kb + c4 * 4);
        }
        #pragma unroll
        for (int i = 0; i < (BK * BN) / (4 * GTH); ++i) {
            const int cid = tid + i * GTH;
            const int row = cid >> 4;
            const int c4  = cid & 15;
            async_copy_b128(&Bs[buf][row * BN + c4 * 4],
                            B + (size_t)(kb + row) * N + tn0 + c4 * 4);
        }
    };

    const int r  = lane & 15;          // M-row (A) / N-col (B) within tile
    const int kk = (lane >> 4) * 2;    // half-wave k offset: 0 or 2

    v8f acc[4] = {v8f{}, v8f{}, v8f{}, v8f{}};

    stage(0, 0);
    wait_async0();
    __syncthreads();

    int buf = 0;
    for (int kb = 0; kb < K; kb += BK) {
        if (kb + BK < K) stage(buf ^ 1, kb + BK);   // overlap DMA with compute

        const float* Arow = &As[buf][(wave * 16 + r) * BK + kk];
        const float* Bbuf = &Bs[buf][0];
        #pragma unroll
        for (int k = 0; k < BK; k += 4) {
            v2f a;                                   // A 16x4: K=k+kk, k+kk+1
            a.x = Arow[k + 0];
            a.y = Arow[k + 1];
            #pragma unroll
            for (int tI = 0; tI < 4; ++tI) {
                v2f b;                               // B 4x16 fragment
                b.x = Bbuf[(k + kk) * BN + tI * 16 + r];
                b.y = Bbuf[(k + kk + 1) * BN + tI * 16 + r];
                acc[tI] = __builtin_amdgcn_wmma_f32_16x16x4_f32(
                    false, a, false, b, (short)0, acc[tI], false, false);
            }
        }

        wait_async0();
        __syncthreads();
        buf ^= 1;
    }

    const int row0 = tm0 + wave * 16 + (lane >> 4) * 8;
    const int col  = lane & 15;
    #pragma unroll
    for (int tI = 0; tI < 4; ++tI) {
        const int cn = tn0 + tI * 16 + col;
        const float bv = bias ? bias[cn] : 0.0f;
        #pragma unroll
        for (int rr = 0; rr < 8; ++rr) {
            C[(size_t)(row0 + rr) * N + cn] = acc[tI][rr] + bv;
        }
    }
}

// ---------------------------------------------------------------------------
// Persistent recurrence: h_t = tanh(xW_t + h_{t-1} @ Wh_h + bh)
// Each block pins a 2048x8 column slice of Wh_h (64KB) in LDS for the whole
// rollout, staged with one TENSOR_LOAD_TO_LDS 2D-tile DMA (TDM, TENSORcnt).
// LDS layout is tile-row-major: wslice[j*CPB + c].
// ---------------------------------------------------------------------------
__global__ __launch_bounds__(RTH)
void rnn_recurrence(const float* __restrict__ xW, const float* __restrict__ Whh,
                    const float* __restrict__ bh, const float* __restrict__ h0,
                    float* Hs, unsigned int* barrier)
{
    extern __shared__ float wslice[];   // [H_DIM][CPB]

    const int tid   = threadIdx.x;
    const int cbase = blockIdx.x * CPB;
    const int w     = tid >> 5;         // wave id = owned column
    const int lane  = tid & 31;

#if __has_builtin(__builtin_amdgcn_tensor_load_to_lds) && \
    __has_builtin(__builtin_amdgcn_s_wait_tensorcnt)
    // --- TDM path: one 2D-tile DMA per block (tile 8 x 2048 of Wh_h) ---
    if (w == 0) {                       // one wave issues (TDM ignores EXEC)
        typedef unsigned int u32x4 __attribute__((ext_vector_type(4)));
        typedef int i32x8 __attribute__((ext_vector_type(8)));
        typedef int i32x4 __attribute__((ext_vector_type(4)));

        const unsigned lds_off = (unsigned)(size_t)(void*)wslice;
        const unsigned long long ga =
            (unsigned long long)(size_t)(const void*)(Whh + cbase);

        // D# group 0: count=1 | lds_addr | global_addr[56:0] | type=2
        u32x4 g0 = { 1u,
                     lds_off,
                     (unsigned)(ga & 0xffffffffull),
                     (unsigned)((ga >> 32) & 0x01ffffffull) | (2u << 30) };
        // D# group 1: data_size=4B; tensor_dim0=H; tensor_dim1=H;
        //             tile_dim0=CPB; tile_dim1=H; tensor_dim0_stride=H
        i32x8 g1 = { (int)(2u << 16),           // [17:16] data_size = 2 (4B)
                     (int)((unsigned)H_DIM << 16),  // [63:48] dim0 lo16
                     (int)((unsigned)H_DIM << 16),  // [79:64]=0, [95:80] dim1 lo16
                     (int)((unsigned)CPB   << 16),  // [111:96]=0, [127:112] tile0
                     (int)H_DIM,                    // [143:128] tile1, [159:144]=0
                     (int)H_DIM,                    // [191:160] dim0_stride lo32
                     0, 0 };
        i32x4 gz4 = { 0, 0, 0, 0 };
        i32x8 gz8 = { 0, 0, 0, 0, 0, 0, 0, 0 };
        __builtin_amdgcn_tensor_load_to_lds(g0, g1, gz4, gz4, gz8, 0);
        __builtin_amdgcn_s_wait_tensorcnt(0);
    }
#else
    // --- fallback: cooperative strided loads, same [j][c] LDS layout ---
    for (int idx = tid; idx < CPB * H_DIM; idx += RTH) {
        const int j = idx >> 3;           // idx / CPB
        const int c = idx & (CPB - 1);    // idx % CPB
        wslice[idx] = Whh[(size_t)j * H_DIM + cbase + c];
    }
#endif
    __syncthreads();

    for (int t = 0; t < T_STEPS; ++t) {
        const float* hprev = (t == 0) ? h0 : (Hs + (size_t)(t - 1) * H_DIM);

        float s = 0.0f;
        #pragma unroll 4
        for (int it = 0; it < H_DIM / 32; ++it) {
            const int j = lane + it * 32;
            s = fmaf(hprev[j], wslice[j * CPB + w], s);
        }
        #pragma unroll
        for (int off = 16; off > 0; off >>= 1)
            s += __shfl_xor(s, off, 32);

        if (lane == 0) {
            const int c = cbase + w;
            const float hv = tanhf(xW[(size_t)t * H_DIM + c] + s + bh[c]);
            Hs[(size_t)t * H_DIM + c] = hv;
        }
        __threadfence();
        __syncthreads();
        if (tid == 0) {
            atomicAdd(barrier, 1u);
            const unsigned target = (unsigned)(t + 1) * NWG;
            while (__hip_atomic_load(barrier, __ATOMIC_ACQUIRE,
                                     __HIP_MEMORY_SCOPE_AGENT) < target) {
                __builtin_amdgcn_s_sleep(1);
            }
        }
        __syncthreads();
    }
}

__global__ void init_barrier(unsigned int* b)
{
    if (threadIdx.x == 0 && blockIdx.x == 0) *b = 0u;
}

__global__ void copy_hfinal(const float* __restrict__ Hs, float* __restrict__ out)
{
    const int i = blockIdx.x * blockDim.x + threadIdx.x;
    if (i < H_DIM)
        out[(size_t)T_STEPS * Y_DIM + i] = Hs[(size_t)(T_STEPS - 1) * H_DIM + i];
}

// ---------------------------------------------------------------------------
extern "C" void kernel_launch(void* const* d_in, const int* in_sizes, int n_in,
                              void* d_out, int out_size, void* d_ws, size_t ws_size,
                              hipStream_t stream)
{
    const float* x   = (const float*)d_in[0];  // [1, T, X]
    const float* h0  = (const float*)d_in[1];  // [H]
    const float* Wx  = (const float*)d_in[2];  // [X, H]
    const float* Whh = (const float*)d_in[3];  // [H, H]
    const float* Why = (const float*)d_in[4];  // [H, Y]
    const float* bh  = (const float*)d_in[5];  // [H]
    const float* by  = (const float*)d_in[6];  // [Y]
    float* out = (float*)d_out;                // ys [T*Y] ++ h_final [H]

    char* ws = (char*)d_ws;
    float* xW = (float*)ws;                                             // 64 MB
    float* Hs = (float*)(ws + (size_t)T_STEPS * H_DIM * sizeof(float)); // 64 MB
    unsigned int* barrier =
        (unsigned int*)(ws + 2 * (size_t)T_STEPS * H_DIM * sizeof(float));

    // 0) reset grid barrier (deterministic per call / per graph replay)
    init_barrier<<<1, 32, 0, stream>>>(barrier);

    // 1) xW = x @ Wx_X   (M=8192, N=2048, K=1024), WMMA f32 + async LDS DMA
    {
        const int blocks = (T_STEPS / BM) * (H_DIM / BN);   // 2048
        gemm_f32_wmma<<<blocks, GTH, 0, stream>>>(
            x, Wx, nullptr, xW, T_STEPS, H_DIM, X_DIM);
    }

    // 2) sequential recurrence, Wh_h resident in LDS across 256 WGPs (TDM-staged)
    {
        const size_t lds_bytes = (size_t)CPB * H_DIM * sizeof(float); // 64 KB
        rnn_recurrence<<<NWG, RTH, lds_bytes, stream>>>(
            xW, Whh, bh, h0, Hs, barrier);
    }

    // 3) ys = Hs @ Wh_y + by   (M=8192, N=1024, K=2048), WMMA f32 -> d_out
    {
        const int blocks = (T_STEPS / BM) * (Y_DIM / BN);   // 1024
        gemm_f32_wmma<<<blocks, GTH, 0, stream>>>(
            Hs, Why, by, out, T_STEPS, Y_DIM, H_DIM);
    }

    // 4) append h_final
    copy_hfinal<<<(H_DIM + 255) / 256, 256, 0, stream>>>(Hs, out);
}